// MessagePassing_463856468618
// MI455X (gfx1250) — compile-verified
//
#include <hip/hip_runtime.h>

#define N_NODES 100000
#define N_EDGES 1250000
#define D_FEAT  64

// 16 lanes per edge: lane c handles features [4c, 4c+4).
// Gather x[col[e]] as float4 (global_load_b128), scatter-add into out[row[e]]
// with a clause of four device-scope hardware f32 atomics.
__global__ __launch_bounds__(256) void MP_scatter_add_kernel(
    const int* __restrict__ row,      // edge_index[0, :]
    const int* __restrict__ col,      // edge_index[1, :]
    const float4* __restrict__ x4,    // x viewed as [N_NODES, 16] float4
    float* __restrict__ out)          // [N_NODES, 64] f32
{
    int tid = blockIdx.x * blockDim.x + threadIdx.x;
    int e = tid >> 4;          // edge id
    int c = tid & 15;          // float4 chunk within the 64-wide feature row
    if (e >= N_EDGES) return;

    // Edge list is streamed exactly once -> non-temporal loads keep WGP$/L2
    // capacity for the reused x/out working sets.
    int r = __builtin_nontemporal_load(row + e);   // destination node
    int s = __builtin_nontemporal_load(col + e);   // source node

    float4 v = x4[(long long)s * 16 + c];          // b128 gather (L2-resident, RT)

    float* dst = out + (long long)r * D_FEAT + c * 4;

    // One address VGPR pair + immediate offsets; clause the 4 atomics so the
    // VMEM pipe services them back-to-back. Device scope => correct cross-WGP
    // reduction, hardware f32 atomic (no CAS loop).
    asm volatile(
        "s_clause 0x3\n\t"
        "global_atomic_add_f32 %0, %1, off scope:SCOPE_DEV\n\t"
        "global_atomic_add_f32 %0, %2, off offset:4 scope:SCOPE_DEV\n\t"
        "global_atomic_add_f32 %0, %3, off offset:8 scope:SCOPE_DEV\n\t"
        "global_atomic_add_f32 %0, %4, off offset:12 scope:SCOPE_DEV"
        :
        : "v"(dst), "v"(v.x), "v"(v.y), "v"(v.z), "v"(v.w)
        : "memory");
}

// Zero the output accumulator. out has N_NODES*D_FEAT floats = 1.6M float4.
// Must run at the start of every launch: the harness graph-replays without
// re-poisoning, so the accumulator must be cleared each call.
__global__ __launch_bounds__(256) void MP_zero_kernel(float4* __restrict__ out, int n4) {
    int i = blockIdx.x * blockDim.x + threadIdx.x;
    if (i < n4) {
        float4 z; z.x = 0.f; z.y = 0.f; z.z = 0.f; z.w = 0.f;
        out[i] = z;
    }
}

extern "C" void kernel_launch(void* const* d_in, const int* in_sizes, int n_in,
                              void* d_out, int out_size, void* d_ws, size_t ws_size,
                              hipStream_t stream) {
    const int*   edge_index = (const int*)d_in[0];        // [2, N_EDGES] int32
    const float* x          = (const float*)d_in[1];      // [N_NODES, 64] f32
    float*       out        = (float*)d_out;              // [N_NODES, 64] f32

    const int* row = edge_index;                 // edge_index[0, :]
    const int* col = edge_index + N_EDGES;       // edge_index[1, :]

    // 1) zero the accumulator
    int n4 = (N_NODES * D_FEAT) / 4;
    int zb = (n4 + 255) / 256;
    MP_zero_kernel<<<zb, 256, 0, stream>>>((float4*)out, n4);

    // 2) scatter-add: 16 threads per edge
    long long total = (long long)N_EDGES * 16;
    int sb = (int)((total + 255) / 256);
    MP_scatter_add_kernel<<<sb, 256, 0, stream>>>(row, col, (const float4*)x, out);
}